// RGATLayer_54082228191481
// MI455X (gfx1250) — compile-verified
//
#include <hip/hip_runtime.h>
#include <math.h>

#define N_NODES 50000
#define N_EDGES 800000
#define IN_DIM  128
#define OUT_DIM 64
#define SLOPE   0.01f

typedef __attribute__((ext_vector_type(2))) float v2f;
typedef __attribute__((ext_vector_type(8))) float v8f;

// pointer types matching the async-LDS builtin prototype:
//   (<4 x i32> addrspace(1)*, <4 x i32> addrspace(3)*, int offset, int cpol)
typedef int v4i __attribute__((vector_size(16)));
typedef __attribute__((address_space(1))) v4i* g_v4i_p;
typedef __attribute__((address_space(3))) v4i* l_v4i_p;

// ---- async global->LDS 16-byte copy (ASYNCcnt-tracked) ----
__device__ __forceinline__ void async_g2l_b128(const void* g, void* l) {
#if __has_builtin(__builtin_amdgcn_global_load_async_to_lds_b128)
    __builtin_amdgcn_global_load_async_to_lds_b128(
        (g_v4i_p)(uintptr_t)g,
        (l_v4i_p)(unsigned)(uintptr_t)l,   // AS3 is 32-bit: low bits = LDS offset
        0, 0);
#else
    unsigned lds_off = (unsigned)(uintptr_t)l;
    unsigned long long ga = (unsigned long long)(uintptr_t)g;
    asm volatile("global_load_async_to_lds_b128 %0, %1, off"
                 :: "v"(lds_off), "v"(ga) : "memory");
#endif
}

__device__ __forceinline__ void wait_async0() {
#if __has_builtin(__builtin_amdgcn_s_wait_asynccnt)
    __builtin_amdgcn_s_wait_asynccnt(0);
#else
    asm volatile("s_wait_asynccnt 0x0" ::: "memory");
#endif
}

// ---- order-preserving float <-> uint encoding for atomicMax-based segment max ----
__device__ __forceinline__ unsigned enc_f32(float f) {
    unsigned b = __float_as_uint(f);
    return (b & 0x80000000u) ? ~b : (b | 0x80000000u);
}
__device__ __forceinline__ float dec_f32(unsigned u) {
    unsigned b = (u & 0x80000000u) ? (u & 0x7FFFFFFFu) : ~u;
    return __uint_as_float(b);
}

// ---------------- init: zero agg/denom, sentinel seg_max ----------------
__global__ void rgat_init_kernel(float* __restrict__ agg, float* __restrict__ denom,
                                 unsigned* __restrict__ segmax) {
    size_t i = (size_t)blockIdx.x * blockDim.x + threadIdx.x;
    if (i < (size_t)N_NODES * OUT_DIM) agg[i] = 0.0f;
    if (i < N_NODES) { denom[i] = 0.0f; segmax[i] = 0u; /* maps below every real */ }
}

// ---------------- WMMA GEMM: z[N,64] = h[N,128] @ W_fc[64,128]^T ----------------
// Block = 4 waves, all sharing row-tile tm = blockIdx.x; wave w handles tn = w.
// The shared 16x128 A-tile (8KB) is DMA'd into LDS once per block with
// GLOBAL_LOAD_ASYNC_TO_LDS_B128, then A-fragments come from LDS (ds_load_b64).
__global__ void rgat_gemm_z_kernel(const float* __restrict__ h,
                                   const float* __restrict__ Wfc,
                                   float* __restrict__ z) {
    __shared__ float sA[16 * IN_DIM];                       // 8 KB

    const int tid  = threadIdx.x;                           // 0..127
    const int tm   = blockIdx.x;                            // 0..3124
    const int tn   = tid >> 5;                              // wave id in block = column tile
    const int lane = tid & 31;
    const int half = lane >> 4;                             // 0 or 1
    const int idx  = lane & 15;                             // M row for A; N col for B/C

    // ---- stage A tile: 4 passes x (128 threads x 16B) = 8KB, coalesced 2KB sweeps ----
    {
        const char* gbase = (const char*)(h + (size_t)tm * 16 * IN_DIM);
        char*       lbase = (char*)sA;
        #pragma unroll
        for (int i = 0; i < 4; ++i) {
            int byte = i * 2048 + tid * 16;
            async_g2l_b128(gbase + byte, lbase + byte);
        }
        wait_async0();
        __syncthreads();
    }

    const float* arow = sA  + (size_t)idx * IN_DIM;               // A row M=idx (LDS)
    const float* brow = Wfc + (size_t)(tn * 16 + idx) * IN_DIM;   // B[k][n] = Wfc[n][k]

    v8f c = {};
    #pragma unroll
    for (int k0 = 0; k0 < IN_DIM; k0 += 4) {
        v2f a, b;
        // A 16x4 layout: VGPR0 = K(2*half), VGPR1 = K(2*half+1)
        a.x = arow[k0 + 2 * half];
        a.y = arow[k0 + 2 * half + 1];
        // B 4x16 layout: VGPR v holds row K = v + 2*half across lanes 0-15 / 16-31
        b.x = brow[k0 + 2 * half];
        b.y = brow[k0 + 2 * half + 1];
        c = __builtin_amdgcn_wmma_f32_16x16x4_f32(false, a, false, b,
                                                  (short)0, c, false, false);
    }
    // C/D 16x16: VGPR v = row M = v + 8*half, col N = idx
    float* zt = z + (size_t)(tm * 16 + 8 * half) * OUT_DIM + tn * 16 + idx;
    #pragma unroll
    for (int v = 0; v < 8; ++v) zt[(size_t)v * OUT_DIM] = c[v];
}

// ---------------- WMMA GEMM: zl[N,64] = z[N,64] @ loop_weight[64,64] ----------------
__global__ void rgat_gemm_loop_kernel(const float* __restrict__ z,
                                      const float* __restrict__ Lw,
                                      float* __restrict__ zl) {
    const int tiles_n = OUT_DIM / 16;                       // 4
    const int wave = blockIdx.x * (blockDim.x >> 5) + (threadIdx.x >> 5);
    const int tm = wave / tiles_n;
    const int tn = wave % tiles_n;
    const int lane = threadIdx.x & 31;
    const int half = lane >> 4;
    const int idx  = lane & 15;

    const float* arow = z + (size_t)(tm * 16 + idx) * OUT_DIM;

    v8f c = {};
    #pragma unroll
    for (int k0 = 0; k0 < OUT_DIM; k0 += 4) {
        v2f a, b;
        a.x = arow[k0 + 2 * half];
        a.y = arow[k0 + 2 * half + 1];
        // B[k][n] = Lw[k*64 + n], row-major
        b.x = Lw[(size_t)(k0 + 2 * half)     * OUT_DIM + tn * 16 + idx];
        b.y = Lw[(size_t)(k0 + 2 * half + 1) * OUT_DIM + tn * 16 + idx];
        c = __builtin_amdgcn_wmma_f32_16x16x4_f32(false, a, false, b,
                                                  (short)0, c, false, false);
    }
    float* zt = zl + (size_t)(tm * 16 + 8 * half) * OUT_DIM + tn * 16 + idx;
    #pragma unroll
    for (int v = 0; v < 8; ++v) zt[(size_t)v * OUT_DIM] = c[v];
}

// ---------------- per-node attention projections s1 = z.w1, s2 = z.w2 ----------------
__global__ void rgat_s12_kernel(const float* __restrict__ z,
                                const float* __restrict__ attn_w,
                                float* __restrict__ s1, float* __restrict__ s2) {
    int i = blockIdx.x * blockDim.x + threadIdx.x;
    if (i >= N_NODES) return;
    const float* zr = z + (size_t)i * OUT_DIM;
    float a1 = 0.0f, a2 = 0.0f;
    #pragma unroll
    for (int d = 0; d < OUT_DIM; d += 4) {
        float4 zv = *(const float4*)(zr + d);
        float4 w1 = *(const float4*)(attn_w + d);
        float4 w2 = *(const float4*)(attn_w + OUT_DIM + d);
        a1 += zv.x * w1.x + zv.y * w1.y + zv.z * w1.z + zv.w * w1.w;
        a2 += zv.x * w2.x + zv.y * w2.y + zv.z * w2.z + zv.w * w2.w;
    }
    s1[i] = a1; s2[i] = a2;
}

// ---------------- per-edge score + leaky relu + segment max ----------------
__global__ void rgat_edge_score_kernel(const float* __restrict__ r_h,
                                       const float* __restrict__ attn_w,
                                       const float* __restrict__ s1,
                                       const float* __restrict__ s2,
                                       const int* __restrict__ esrc,
                                       const int* __restrict__ edst,
                                       float* __restrict__ a_e,
                                       unsigned* __restrict__ segmax) {
    int e = blockIdx.x * blockDim.x + threadIdx.x;
    if (e >= N_EDGES) return;
    const float* rr = r_h + (size_t)e * OUT_DIM;
    const float* w3 = attn_w + 2 * OUT_DIM;
    // streaming prefetch of a future r_h row (global_prefetch_b8)
    __builtin_prefetch(rr + (size_t)256 * OUT_DIM, 0, 1);
    float acc = 0.0f;
    #pragma unroll
    for (int d = 0; d < OUT_DIM; d += 4) {
        float4 rv = *(const float4*)(rr + d);
        float4 wv = *(const float4*)(w3 + d);
        acc += rv.x * wv.x + rv.y * wv.y + rv.z * wv.z + rv.w * wv.w;
    }
    int s = esrc[e], d = edst[e];
    float a = acc + s1[s] + s2[d];
    float el = (a > 0.0f) ? a : SLOPE * a;      // leaky relu
    a_e[e] = el;
    atomicMax(&segmax[d], enc_f32(el));
}

// ---------------- per-edge exp(e - segmax[dst]) + denom ----------------
__global__ void rgat_edge_exp_kernel(const int* __restrict__ edst,
                                     const unsigned* __restrict__ segmax,
                                     float* __restrict__ a_e,
                                     float* __restrict__ denom) {
    int e = blockIdx.x * blockDim.x + threadIdx.x;
    if (e >= N_EDGES) return;
    int d = edst[e];
    float m = dec_f32(segmax[d]);
    float ex = __expf(a_e[e] - m);
    a_e[e] = ex;                                 // in-place: a_e now holds e_exp
    atomicAdd(&denom[d], ex);
}

// ---------------- wave-per-edge weighted scatter: agg[dst] += alpha * z[src] ----------------
__global__ void rgat_edge_agg_kernel(const int* __restrict__ esrc,
                                     const int* __restrict__ edst,
                                     const float* __restrict__ e_exp,
                                     const float* __restrict__ denom,
                                     const float* __restrict__ z,
                                     float* __restrict__ agg) {
    int wid  = (blockIdx.x * blockDim.x + threadIdx.x) >> 5;   // one wave32 per edge
    int lane = threadIdx.x & 31;
    if (wid >= N_EDGES) return;
    int s = esrc[wid], d = edst[wid];
    float alpha = e_exp[wid] / denom[d];
    float2 zv = *(const float2*)(z + (size_t)s * OUT_DIM + 2 * lane);  // L2 hit
    atomicAdd(&agg[(size_t)d * OUT_DIM + 2 * lane],     alpha * zv.x);
    atomicAdd(&agg[(size_t)d * OUT_DIM + 2 * lane + 1], alpha * zv.y);
}

// ---------------- h_out = where(denom>0, relu(agg + z@loop_weight), 0) ----------------
__global__ void rgat_final_kernel(const float* __restrict__ agg,
                                  const float* __restrict__ zl,
                                  const float* __restrict__ denom,
                                  float* __restrict__ out) {
    size_t i = (size_t)blockIdx.x * blockDim.x + threadIdx.x;
    if (i >= (size_t)N_NODES * OUT_DIM) return;
    int row = (int)(i / OUT_DIM);
    float v = agg[i] + zl[i];
    v = (v > 0.0f) ? v : 0.0f;
    out[i] = (denom[row] > 0.0f) ? v : 0.0f;
}

extern "C" void kernel_launch(void* const* d_in, const int* in_sizes, int n_in,
                              void* d_out, int out_size, void* d_ws, size_t ws_size,
                              hipStream_t stream) {
    const float* h      = (const float*)d_in[0];   // [N,128]
    const float* r_h    = (const float*)d_in[1];   // [E,64]
    const float* W_fc   = (const float*)d_in[2];   // [64,128]
    const float* attn_w = (const float*)d_in[3];   // [1,192]
    const float* loop_w = (const float*)d_in[4];   // [64,64]
    const int*   esrc   = (const int*)d_in[5];     // [E]
    const int*   edst   = (const int*)d_in[6];     // [E]
    float* out = (float*)d_out;                    // [N,64]

    // workspace carve-up
    char* p = (char*)d_ws;
    float*    z      = (float*)p;    p += sizeof(float) * (size_t)N_NODES * OUT_DIM;
    float*    zl     = (float*)p;    p += sizeof(float) * (size_t)N_NODES * OUT_DIM;
    float*    agg    = (float*)p;    p += sizeof(float) * (size_t)N_NODES * OUT_DIM;
    float*    a_e    = (float*)p;    p += sizeof(float) * (size_t)N_EDGES;
    float*    denom  = (float*)p;    p += sizeof(float) * (size_t)N_NODES;
    float*    s1     = (float*)p;    p += sizeof(float) * (size_t)N_NODES;
    float*    s2     = (float*)p;    p += sizeof(float) * (size_t)N_NODES;
    unsigned* segmax = (unsigned*)p; p += sizeof(unsigned) * (size_t)N_NODES;

    const int elems = N_NODES * OUT_DIM;           // 3.2M

    // 1. init
    rgat_init_kernel<<<(elems + 255) / 256, 256, 0, stream>>>(agg, denom, segmax);

    // 2. z = h @ W_fc^T  (WMMA + async LDS staging of shared A tile; 1 block per row tile)
    rgat_gemm_z_kernel<<<N_NODES / 16, 128, 0, stream>>>(h, W_fc, z);

    // 3. zl = z @ loop_weight (WMMA)
    {
        int waves = (N_NODES / 16) * (OUT_DIM / 16);  // 12500
        rgat_gemm_loop_kernel<<<waves / 4, 128, 0, stream>>>(z, loop_w, zl);
    }

    // 4. s1/s2
    rgat_s12_kernel<<<(N_NODES + 255) / 256, 256, 0, stream>>>(z, attn_w, s1, s2);

    // 5. edge scores + seg max
    rgat_edge_score_kernel<<<(N_EDGES + 255) / 256, 256, 0, stream>>>(
        r_h, attn_w, s1, s2, esrc, edst, a_e, segmax);

    // 6. exp + denom
    rgat_edge_exp_kernel<<<(N_EDGES + 255) / 256, 256, 0, stream>>>(
        edst, segmax, a_e, denom);

    // 7. weighted scatter (wave per edge)
    rgat_edge_agg_kernel<<<(N_EDGES * 32 + 255) / 256, 256, 0, stream>>>(
        esrc, edst, a_e, denom, z, agg);

    // 8. final relu + mask
    rgat_final_kernel<<<(elems + 255) / 256, 256, 0, stream>>>(agg, zl, denom, out);
}